// MoE_80556406603831
// MI455X (gfx1250) — compile-verified
//
#include <hip/hip_runtime.h>
#include <hip/hip_bf16.h>
#include <math.h>

#define T_TOK   2048
#define E_DIM   512
#define H_HEADS 8
#define D_HEAD  64
#define HID_DIM 2048
#define NE_EXP  4
#define NEG_VAL -1000000000.0f

typedef __attribute__((ext_vector_type(2))) float v2f;
typedef __attribute__((ext_vector_type(8))) float v8f;

// D = A(16x4) * B(4x16) + C, fp32 matrix pipe (full reference precision)
__device__ inline v8f wmma_f32(v2f a, v2f b, v8f c) {
    return __builtin_amdgcn_wmma_f32_16x16x4_f32(false, a, false, b, (short)0, c, false, false);
}

// ---------------------------------------------------------------- zero output
__global__ void zero_out_kernel(float* __restrict__ out, int nElem) {
    int i = blockIdx.x * blockDim.x + threadIdx.x;
    if (i < nElem) out[i] = 0.0f;
}

// ---------------------------------------------------------------- gating: logits, top-2, softmax
__global__ __launch_bounds__(32) void gate_kernel(
    const float* __restrict__ xf, const float* __restrict__ gw,
    const float* __restrict__ gb, int* __restrict__ sel, float* __restrict__ probs)
{
    int t = blockIdx.x;
    int lane = threadIdx.x;
    const float* xr = xf + (size_t)t * E_DIM;
    float acc[NE_EXP] = {0.f, 0.f, 0.f, 0.f};
    for (int j = lane; j < E_DIM; j += 32) {
        float xv = xr[j];
#pragma unroll
        for (int e = 0; e < NE_EXP; ++e) acc[e] += xv * gw[e * E_DIM + j];
    }
#pragma unroll
    for (int e = 0; e < NE_EXP; ++e)
        for (int off = 16; off > 0; off >>= 1)
            acc[e] += __shfl_xor(acc[e], off, 32);
    if (lane == 0) {
        float lg[NE_EXP];
#pragma unroll
        for (int e = 0; e < NE_EXP; ++e) lg[e] = acc[e] + gb[e];
        int e0 = 0;
        for (int e = 1; e < NE_EXP; ++e) if (lg[e] > lg[e0]) e0 = e;   // ties -> lowest idx
        int e1 = -1;
        for (int e = 0; e < NE_EXP; ++e) {
            if (e == e0) continue;
            if (e1 < 0 || lg[e] > lg[e1]) e1 = e;
        }
        float ex = expf(lg[e1] - lg[e0]);
        float p0 = 1.0f / (1.0f + ex);
        sel[t] = e0;          sel[T_TOK + t] = e1;
        probs[t] = p0;        probs[T_TOK + t] = 1.0f - p0;
    }
}

// ---------------------------------------------------------------- deterministic list build (no atomics)
__global__ void build_lists_kernel(const int* __restrict__ sel,
                                   int* __restrict__ counts, int* __restrict__ lists)
{
    int c = threadIdx.x;
    if (c >= 2 * NE_EXP) return;
    int slot = c >> 2, e = c & 3;
    const int* s = sel + slot * T_TOK;
    int* L = lists + c * T_TOK;
    int cnt = 0;
    for (int t = 0; t < T_TOK; ++t)
        if (s[t] == e) L[cnt++] = t;
    counts[c] = cnt;
}

// ---------------------------------------------------------------- gather routed tokens (clamped pad rows)
__global__ __launch_bounds__(128) void gather_kernel(
    const float* __restrict__ xf, const int* __restrict__ list,
    const int* __restrict__ ntok, float* __restrict__ xg)
{
    int n = *ntok;
    if (n == 0) return;
    int row = blockIdx.x;
    int src = list[min(row, n - 1)];
    const float4* s = (const float4*)(xf + (size_t)src * E_DIM);
    float4* d = (float4*)(xg + (size_t)row * E_DIM);
    d[threadIdx.x] = s[threadIdx.x];   // 128 lanes x float4 = 512 floats
}

// ---------------------------------------------------------------- LayerNorm, one wave per row
__global__ __launch_bounds__(32) void ln_kernel(
    const float* __restrict__ x, const float* __restrict__ w,
    const float* __restrict__ b, float* __restrict__ y)
{
    int row = blockIdx.x;
    int lane = threadIdx.x;
    const float* xr = x + (size_t)row * E_DIM;
    float v[16];
    float s = 0.f;
#pragma unroll
    for (int j = 0; j < 16; ++j) { v[j] = xr[lane + 32 * j]; s += v[j]; }
    for (int off = 16; off > 0; off >>= 1) s += __shfl_xor(s, off, 32);
    float mu = s * (1.0f / E_DIM);
    float var = 0.f;
#pragma unroll
    for (int j = 0; j < 16; ++j) { float d = v[j] - mu; var += d * d; }
    for (int off = 16; off > 0; off >>= 1) var += __shfl_xor(var, off, 32);
    float r = rsqrtf(var * (1.0f / E_DIM) + 1e-5f);
#pragma unroll
    for (int j = 0; j < 16; ++j) {
        int col = lane + 32 * j;
        y[(size_t)row * E_DIM + col] = (v[j] - mu) * r * w[col] + b[col];
    }
}

// ---------------------------------------------------------------- WMMA GEMM  C[M,N] = A[M,K] @ W[N,K]^T + bias
// one wave -> 64x64 C tile (16 independent v8f accumulators, 16 WMMA per 8 b64 loads)
enum { EPI_NONE = 0, EPI_GELU = 1, EPI_RESID = 2, EPI_SCATTER = 3 };

template <int EPI>
__global__ __launch_bounds__(32) void gemm_xwt_kernel(
    const float* __restrict__ A, int lda,
    const float* __restrict__ W,            // [N][K] row-major
    const float* __restrict__ bias,
    float* __restrict__ C, int ldc, int K,
    const int* __restrict__ ntok,
    const float* __restrict__ resid,        // RESID: residual input; SCATTER: x1
    const int* __restrict__ list,           // SCATTER: token indices
    const float* __restrict__ prob,         // SCATTER: routing weight per token
    float* __restrict__ outp)               // SCATTER: output accumulator
{
    int n = *ntok;
    int rowBase = blockIdx.x * 64;
    if (rowBase >= n) return;               // device-count guard keeps graph static but skips work
    int colBase = blockIdx.y * 64;
    int lane  = threadIdx.x;
    int mrow  = lane & 15;
    int khalf = lane >> 4;

    v8f c[4][4] = {};
    const float* ap[4];
    const float* bp[4];
#pragma unroll
    for (int t = 0; t < 4; ++t) {
        ap[t] = A + (size_t)(rowBase + 16 * t + mrow) * lda + 2 * khalf;
        bp[t] = W + (size_t)(colBase + 16 * t + mrow) * K + 2 * khalf;
    }

#pragma unroll 2
    for (int k = 0; k < K; k += 4) {
        v2f af[4], bf[4];
#pragma unroll
        for (int t = 0; t < 4; ++t) {
            af[t] = *(const v2f*)(ap[t] + k);
            bf[t] = *(const v2f*)(bp[t] + k);
        }
#pragma unroll
        for (int i = 0; i < 4; ++i)
#pragma unroll
            for (int j = 0; j < 4; ++j)
                c[i][j] = wmma_f32(af[i], bf[j], c[i][j]);
    }

    int ncol = lane & 15;
#pragma unroll
    for (int ti = 0; ti < 4; ++ti)
#pragma unroll
        for (int tj = 0; tj < 4; ++tj) {
            int col = colBase + 16 * tj + ncol;
            float bv = bias[col];
#pragma unroll
            for (int i = 0; i < 8; ++i) {
                int row = rowBase + 16 * ti + i + 8 * khalf;
                float v = c[ti][tj][i] + bv;
                if (EPI == EPI_GELU)
                    v = 0.5f * v * (1.0f + erff(v * 0.70710678118654752f)); // exact gelu
                if (EPI == EPI_RESID)
                    v += resid[(size_t)row * E_DIM + col];
                if (EPI == EPI_SCATTER) {
                    if (row < n) {
                        int t = list[row];
                        float wt = prob[t];
                        float x1v = resid[(size_t)row * E_DIM + col];
                        outp[(size_t)t * E_DIM + col] += wt * (x1v + v);
                    }
                } else {
                    C[(size_t)row * ldc + col] = v;
                }
            }
        }
}

// ---------------------------------------------------------------- flash attention within routed set
// grid.x = query tiles of 16, grid.y = head; one wave per block
__global__ __launch_bounds__(32) void attn_kernel(
    const float* __restrict__ qkv, float* __restrict__ o, const int* __restrict__ ntok)
{
    __shared__ float smP[16 * 16];
    int n = *ntok;
    int qBase = blockIdx.x * 16;
    if (qBase >= n) return;
    int h = blockIdx.y;
    int lane  = threadIdx.x;
    int mrow  = lane & 15;
    int khalf = lane >> 4;
    int ncol  = lane & 15;

    // Q fragments for 16 k-steps of 4 over D=64
    v2f qf[16];
    const float* qp = qkv + (size_t)(qBase + mrow) * 1536 + h * 64 + 2 * khalf;
#pragma unroll
    for (int s = 0; s < 16; ++s) qf[s] = *(const v2f*)(qp + 4 * s);

    const float* Kb = qkv + 512  + h * 64;
    const float* Vb = qkv + 1024 + h * 64;

    v8f acc[4] = {};
    float rm[8], rl[8];
#pragma unroll
    for (int i = 0; i < 8; ++i) { rm[i] = -3.0e38f; rl[i] = 0.f; }

    for (int s0 = 0; s0 < n; s0 += 16) {
        // S = Q K^T : two independent accumulator chains to break the WMMA RAW chain
        v8f sc0 = {}, sc1 = {};
        const float* kp = Kb + (size_t)(s0 + mrow) * 1536 + 2 * khalf;
#pragma unroll
        for (int d = 0; d < 16; d += 2) {
            v2f kf0 = *(const v2f*)(kp + 4 * d);
            v2f kf1 = *(const v2f*)(kp + 4 * d + 4);
            sc0 = wmma_f32(qf[d],     kf0, sc0);
            sc1 = wmma_f32(qf[d + 1], kf1, sc1);
        }
        v8f sc = sc0 + sc1;

        bool valid = (s0 + ncol) < n;
        float p[8];
#pragma unroll
        for (int i = 0; i < 8; ++i) {
            float sv = valid ? sc[i] * 0.125f : NEG_VAL;   // 1/sqrt(64), key mask
            float tm = sv;
            for (int off = 1; off < 16; off <<= 1) tm = fmaxf(tm, __shfl_xor(tm, off, 32));
            float mn   = fmaxf(rm[i], tm);
            float corr = expf(rm[i] - mn);
            float pv   = expf(sv - mn);
            float ps   = pv;
            for (int off = 1; off < 16; off <<= 1) ps += __shfl_xor(ps, off, 32);
            rl[i] = rl[i] * corr + ps;
            rm[i] = mn;
            p[i]  = pv;
#pragma unroll
            for (int ct = 0; ct < 4; ++ct) acc[ct][i] *= corr;
        }
        // C-layout P -> A-layout through LDS, then O += P V
#pragma unroll
        for (int i = 0; i < 8; ++i) smP[(i + 8 * khalf) * 16 + ncol] = p[i];
        __syncthreads();
#pragma unroll
        for (int st = 0; st < 4; ++st) {
            v2f pa;
            pa.x = smP[mrow * 16 + 4 * st + 2 * khalf];
            pa.y = smP[mrow * 16 + 4 * st + 2 * khalf + 1];
            const float* vp = Vb + (size_t)(s0 + 4 * st + 2 * khalf) * 1536;
#pragma unroll
            for (int ct = 0; ct < 4; ++ct) {
                int dcol = ct * 16 + ncol;
                v2f vb;
                vb.x = vp[dcol];
                vb.y = vp[1536 + dcol];
                acc[ct] = wmma_f32(pa, vb, acc[ct]);
            }
        }
        __syncthreads();
    }
#pragma unroll
    for (int ct = 0; ct < 4; ++ct)
#pragma unroll
        for (int i = 0; i < 8; ++i) {
            int row = qBase + i + 8 * khalf;
            o[(size_t)row * E_DIM + h * 64 + ct * 16 + ncol] = acc[ct][i] / rl[i];
        }
}

// ---------------------------------------------------------------- host orchestration
extern "C" void kernel_launch(void* const* d_in, const int* in_sizes, int n_in,
                              void* d_out, int out_size, void* d_ws, size_t ws_size,
                              hipStream_t stream)
{
    const float* x      = (const float*)d_in[0];
    const float* gate_w = (const float*)d_in[1];
    const float* gate_b = (const float*)d_in[2];
    const float* ln1_w  = (const float*)d_in[3];
    const float* ln1_b  = (const float*)d_in[4];
    const float* ipw    = (const float*)d_in[5];
    const float* ipb    = (const float*)d_in[6];
    const float* opw    = (const float*)d_in[7];
    const float* opb    = (const float*)d_in[8];
    const float* ln2_w  = (const float*)d_in[9];
    const float* ln2_b  = (const float*)d_in[10];
    const float* w1     = (const float*)d_in[11];
    const float* b1     = (const float*)d_in[12];
    const float* w2     = (const float*)d_in[13];
    const float* b2     = (const float*)d_in[14];
    float* out = (float*)d_out;

    float* ws    = (float*)d_ws;
    float* probs = ws;                                   // 2*T floats
    int*   sel   = (int*)(ws + 2 * T_TOK);               // 2*T ints
    int*   counts= (int*)(ws + 4 * T_TOK);               // 8 (padded to 32)
    int*   lists = counts + 32;                          // 8*T ints
    float* xg    = ws + 4 * T_TOK + 32 + 8 * T_TOK;      // T*E  (also h2)
    float* hbuf  = xg + (size_t)T_TOK * E_DIM;           // T*E  (LN1 out, then attn out)
    float* qkv   = hbuf + (size_t)T_TOK * E_DIM;         // T*3E
    float* x1    = qkv + (size_t)T_TOK * 3 * E_DIM;      // T*E
    float* m1    = x1 + (size_t)T_TOK * E_DIM;           // T*HID

    zero_out_kernel<<<(T_TOK * E_DIM + 255) / 256, 256, 0, stream>>>(out, T_TOK * E_DIM);
    gate_kernel<<<T_TOK, 32, 0, stream>>>(x, gate_w, gate_b, sel, probs);
    build_lists_kernel<<<1, 32, 0, stream>>>(sel, counts, lists);

    for (int c = 0; c < 2 * NE_EXP; ++c) {
        int slot = c >> 2, e = c & 3;
        const int* nt = counts + c;
        const int* Lc = lists + c * T_TOK;

        gather_kernel<<<T_TOK, 128, 0, stream>>>(x, Lc, nt, xg);
        ln_kernel<<<T_TOK, 32, 0, stream>>>(xg, ln1_w + e * E_DIM, ln1_b + e * E_DIM, hbuf);

        dim3 g1(T_TOK / 64, (3 * E_DIM) / 64);
        gemm_xwt_kernel<EPI_NONE><<<g1, 32, 0, stream>>>(
            hbuf, E_DIM, ipw + (size_t)e * 3 * E_DIM * E_DIM, ipb + e * 3 * E_DIM,
            qkv, 3 * E_DIM, E_DIM, nt, nullptr, nullptr, nullptr, nullptr);

        dim3 ga(T_TOK / 16, H_HEADS);
        attn_kernel<<<ga, 32, 0, stream>>>(qkv, hbuf, nt);

        dim3 g2(T_TOK / 64, E_DIM / 64);
        gemm_xwt_kernel<EPI_RESID><<<g2, 32, 0, stream>>>(
            hbuf, E_DIM, opw + (size_t)e * E_DIM * E_DIM, opb + e * E_DIM,
            x1, E_DIM, E_DIM, nt, xg, nullptr, nullptr, nullptr);

        ln_kernel<<<T_TOK, 32, 0, stream>>>(x1, ln2_w + e * E_DIM, ln2_b + e * E_DIM, xg);

        dim3 g3(T_TOK / 64, HID_DIM / 64);
        gemm_xwt_kernel<EPI_GELU><<<g3, 32, 0, stream>>>(
            xg, E_DIM, w1 + (size_t)e * HID_DIM * E_DIM, b1 + e * HID_DIM,
            m1, HID_DIM, E_DIM, nt, nullptr, nullptr, nullptr, nullptr);

        dim3 g4(T_TOK / 64, E_DIM / 64);
        gemm_xwt_kernel<EPI_SCATTER><<<g4, 32, 0, stream>>>(
            m1, HID_DIM, w2 + (size_t)e * E_DIM * HID_DIM, b2 + e * E_DIM,
            nullptr, 0, HID_DIM, nt, x1, Lc, probs + slot * T_TOK, out);
    }
}